// GLFP_F_61237643706853
// MI455X (gfx1250) — compile-verified
//
#include <hip/hip_runtime.h>

// ---------------------------------------------------------------------------
// GATv2 (2 layers) + GraphNorm, MI455X / gfx1250.
// - Node GEMMs: V_WMMA_F32_16X16X4_F32 (exact f32 matrix math), branchless
//   inner loop (no per-iteration EXEC masking).
// - Edge passes: fused edge-attr transform + neighborhood softmax using
//   order-preserving-uint atomic max + f32 atomic adds. Edge streams use
//   non-temporal loads so the 192 MB L2 keeps the node-side working set.
// ---------------------------------------------------------------------------

constexpr int N_NODES = 100000;
constexpr int N_EDGES = 1600000;
constexpr int NDIM    = 128;
constexpr int EDIM    = 16;
constexpr int H       = 10;
constexpr int G       = 64;
constexpr int C1V     = 5;   // layer-1 channels/head  -> H*C1 = 50
constexpr float SLOPE = 0.2f;
constexpr float EPSV  = 1e-5f;

typedef __attribute__((ext_vector_type(2))) float v2f;
typedef __attribute__((ext_vector_type(4))) float v4f;
typedef __attribute__((ext_vector_type(8))) float v8f;

__device__ __forceinline__ unsigned f2ord(float f) {
  unsigned u = __float_as_uint(f);
  return (u & 0x80000000u) ? ~u : (u | 0x80000000u);
}
__device__ __forceinline__ float ord2f(unsigned u) {
  return (u & 0x80000000u) ? __uint_as_float(u & 0x7FFFFFFFu)
                           : __uint_as_float(~u);
}

// ---------------------------------------------------------------------------
// Layer-1 node transform: xl = x@Wl + bl, xr = x@Wr + br  via f32 WMMA.
// One wave = one 16(M)x16(N) tile; 4 waves/block cover the 50 output cols.
// A fragment (16x4 f32, 2 VGPRs): v0 = K {0 | 2}, v1 = K {1 | 3} per lane-half.
// C/D (16x16 f32, 8 VGPRs): vgpr r, lanes 0-15 -> M=r, lanes 16-31 -> M=r+8.
// B loads are clamped + cndmask-zeroed: EXEC stays all-ones in the K-loop.
// ---------------------------------------------------------------------------
__global__ __launch_bounds__(128) void k_node_gemm1(
    const float* __restrict__ x,
    const float* __restrict__ wl, const float* __restrict__ bl,
    const float* __restrict__ wr, const float* __restrict__ br,
    float* __restrict__ xl, float* __restrict__ xr) {
  const int lane = threadIdx.x & 31;
  const int wid  = threadIdx.x >> 5;          // n-tile 0..3
  const int m0   = blockIdx.x * 16;           // 6250 * 16 == 100000 exactly
  const int n0   = wid * 16;
  const int l15  = lane & 15;
  const int hi   = lane >> 4;                 // 0 or 1 (lane half)
  const int row  = m0 + l15;
  const int rowc = row < N_NODES ? row : N_NODES - 1;
  const int col  = n0 + l15;
  const bool cok = col < H * C1V;
  const int colc = cok ? col : (H * C1V - 1); // in-bounds clamp

  v8f accl = {}; v8f accr = {};
  #pragma unroll 4
  for (int kk = 0; kk < NDIM / 4; ++kk) {
    const int k = kk * 4 + hi * 2;
    v2f a;
    a.x = x[(size_t)rowc * NDIM + k];
    a.y = x[(size_t)rowc * NDIM + k + 1];
    v2f bvl, bvr;
    bvl.x = wl[k * (H * C1V) + colc];
    bvl.y = wl[(k + 1) * (H * C1V) + colc];
    bvr.x = wr[k * (H * C1V) + colc];
    bvr.y = wr[(k + 1) * (H * C1V) + colc];
    bvl.x = cok ? bvl.x : 0.f;  bvl.y = cok ? bvl.y : 0.f;   // v_cndmask
    bvr.x = cok ? bvr.x : 0.f;  bvr.y = cok ? bvr.y : 0.f;
    accl = __builtin_amdgcn_wmma_f32_16x16x4_f32(false, a, false, bvl,
                                                 (short)0, accl, false, false);
    accr = __builtin_amdgcn_wmma_f32_16x16x4_f32(false, a, false, bvr,
                                                 (short)0, accr, false, false);
  }
  if (cok) {
    const float addl = bl[col];
    const float addr = br[col];
    #pragma unroll
    for (int r = 0; r < 8; ++r) {
      const int m = m0 + r + 8 * hi;
      if (m < N_NODES) {
        xl[(size_t)m * (H * C1V) + col] = accl[r] + addl;
        xr[(size_t)m * (H * C1V) + col] = accr[r] + addr;
      }
    }
  }
}

// ---------------------------------------------------------------------------
// Per-edge logits (shared by max-pass and accumulate-pass, CH = channels/head)
// edge_attr is streamed with NT hint (use-once, keep L2 for node arrays).
// ---------------------------------------------------------------------------
template <int CH>
__device__ __forceinline__ void compute_logits(
    int e, int s, int d,
    const float* __restrict__ eattr, const float* __restrict__ xl,
    const float* __restrict__ xr, const float* __restrict__ w_e,
    const float* __restrict__ att, float (&lg)[H]) {
  const v4f* ep = (const v4f*)(eattr + (size_t)e * EDIM);  // 64B aligned
  v4f ev[EDIM / 4];
  #pragma unroll
  for (int q = 0; q < EDIM / 4; ++q)
    ev[q] = __builtin_nontemporal_load(ep + q);            // th: NT

  float acc[H * CH];
  #pragma unroll
  for (int j = 0; j < H * CH; ++j) acc[j] = 0.f;
  for (int i = 0; i < EDIM; ++i) {          // w_e indices lane-uniform -> SMEM
    const float av = ev[i >> 2][i & 3];
    #pragma unroll
    for (int j = 0; j < H * CH; ++j)
      acc[j] = fmaf(av, w_e[i * H * CH + j], acc[j]);
  }
  const float* xls = xl + (size_t)s * H * CH;   // L2-resident gathers
  const float* xrd = xr + (size_t)d * H * CH;
  #pragma unroll
  for (int hh = 0; hh < H; ++hh) {
    float l = 0.f;
    #pragma unroll
    for (int c = 0; c < CH; ++c) {
      const int j = hh * CH + c;
      float m = acc[j] + xls[j] + xrd[j];
      m = m > 0.f ? m : SLOPE * m;           // leaky_relu(0.2)
      l = fmaf(m, att[j], l);
    }
    lg[hh] = l;
  }
}

template <int CH>
__global__ __launch_bounds__(256) void k_edge_max(
    const int* __restrict__ ei, const float* __restrict__ eattr,
    const float* __restrict__ xl, const float* __restrict__ xr,
    const float* __restrict__ w_e, const float* __restrict__ att,
    unsigned* __restrict__ lmax) {
  const int e = blockIdx.x * blockDim.x + threadIdx.x;
  if (e >= N_EDGES) return;
  const int s = __builtin_nontemporal_load(ei + e);
  const int d = __builtin_nontemporal_load(ei + N_EDGES + e);
  float lg[H];
  compute_logits<CH>(e, s, d, eattr, xl, xr, w_e, att, lg);
  #pragma unroll
  for (int hh = 0; hh < H; ++hh)
    atomicMax(&lmax[(size_t)d * H + hh], f2ord(lg[hh]));
}

template <int CH>
__global__ __launch_bounds__(256) void k_edge_acc(
    const int* __restrict__ ei, const float* __restrict__ eattr,
    const float* __restrict__ xl, const float* __restrict__ xr,
    const float* __restrict__ w_e, const float* __restrict__ att,
    const unsigned* __restrict__ lmax,
    float* __restrict__ denom, float* __restrict__ outnum) {
  const int e = blockIdx.x * blockDim.x + threadIdx.x;
  if (e >= N_EDGES) return;
  const int s = __builtin_nontemporal_load(ei + e);
  const int d = __builtin_nontemporal_load(ei + N_EDGES + e);
  float lg[H];
  compute_logits<CH>(e, s, d, eattr, xl, xr, w_e, att, lg);
  const float* xls = xl + (size_t)s * H * CH;
  #pragma unroll
  for (int hh = 0; hh < H; ++hh) {
    const float lm = ord2f(lmax[(size_t)d * H + hh]);
    const float pe = __expf(lg[hh] - lm);
    atomicAdd(&denom[(size_t)d * H + hh], pe);
    #pragma unroll
    for (int c = 0; c < CH; ++c)
      atomicAdd(&outnum[(size_t)d * H * CH + hh * CH + c],
                pe * xls[hh * CH + c]);
  }
}

// ---------------------------------------------------------------------------
// Layer-1 finalize: head-mean + bias, and start GraphNorm reductions
// ---------------------------------------------------------------------------
__global__ __launch_bounds__(256) void k_fin1(
    const float* __restrict__ outnum, const float* __restrict__ denom,
    const float* __restrict__ bias1, const int* __restrict__ batch,
    float* __restrict__ y1, float* __restrict__ gn_sum,
    float* __restrict__ gn_cnt) {
  const int n = blockIdx.x * blockDim.x + threadIdx.x;
  if (n >= N_NODES) return;
  const int g = batch[n];
  float inv[H];
  #pragma unroll
  for (int hh = 0; hh < H; ++hh) {
    const float dn = denom[(size_t)n * H + hh];
    inv[hh] = dn > 0.f ? 1.f / dn : 0.f;    // no-in-edge node -> 0
  }
  #pragma unroll
  for (int c = 0; c < C1V; ++c) {
    float s = 0.f;
    #pragma unroll
    for (int hh = 0; hh < H; ++hh)
      s = fmaf(outnum[(size_t)n * H * C1V + hh * C1V + c], inv[hh], s);
    const float v = s * (1.f / H) + bias1[c];
    y1[(size_t)n * C1V + c] = v;
    atomicAdd(&gn_sum[g * C1V + c], v);
  }
  atomicAdd(&gn_cnt[g], 1.f);
}

__global__ void k_gn_mean(const float* __restrict__ gn_sum,
                          const float* __restrict__ gn_cnt,
                          float* __restrict__ gn_mean) {
  const int i = blockIdx.x * blockDim.x + threadIdx.x;
  if (i >= G * C1V) return;
  gn_mean[i] = gn_sum[i] / fmaxf(gn_cnt[i / C1V], 1.f);
}

__global__ __launch_bounds__(256) void k_gn_center(
    const int* __restrict__ batch, const float* __restrict__ gn_mean,
    const float* __restrict__ gn_ms, float* __restrict__ y1,
    float* __restrict__ gn_var) {
  const int n = blockIdx.x * blockDim.x + threadIdx.x;
  if (n >= N_NODES) return;
  const int g = batch[n];
  #pragma unroll
  for (int c = 0; c < C1V; ++c) {
    const float xc =
        y1[(size_t)n * C1V + c] - gn_mean[g * C1V + c] * gn_ms[c];
    y1[(size_t)n * C1V + c] = xc;           // store centered value in place
    atomicAdd(&gn_var[g * C1V + c], xc * xc);
  }
}

__global__ void k_gn_rstd(const float* __restrict__ gn_var,
                          const float* __restrict__ gn_cnt,
                          float* __restrict__ gn_rstd) {
  const int i = blockIdx.x * blockDim.x + threadIdx.x;
  if (i >= G * C1V) return;
  gn_rstd[i] = rsqrtf(gn_var[i] / fmaxf(gn_cnt[i / C1V], 1.f) + EPSV);
}

__global__ __launch_bounds__(256) void k_gn_apply(
    const int* __restrict__ batch, const float* __restrict__ y1,
    const float* __restrict__ gn_rstd, const float* __restrict__ gn_w,
    const float* __restrict__ gn_b, float* __restrict__ h1) {
  const int n = blockIdx.x * blockDim.x + threadIdx.x;
  if (n >= N_NODES) return;
  const int g = batch[n];
  #pragma unroll
  for (int c = 0; c < C1V; ++c) {
    const float v = gn_w[c] * y1[(size_t)n * C1V + c] * gn_rstd[g * C1V + c] +
                    gn_b[c];
    h1[(size_t)n * C1V + c] = fmaxf(v, 0.f);  // relu
  }
}

// ---------------------------------------------------------------------------
// Layer-2 node transform: tiny K=5 GEMV per node (VALU)
// ---------------------------------------------------------------------------
__global__ __launch_bounds__(256) void k_lin2(
    const float* __restrict__ h1, const float* __restrict__ w2l,
    const float* __restrict__ b2l, const float* __restrict__ w2r,
    const float* __restrict__ b2r, float* __restrict__ xl2,
    float* __restrict__ xr2) {
  const int n = blockIdx.x * blockDim.x + threadIdx.x;
  if (n >= N_NODES) return;
  float hv[C1V];
  #pragma unroll
  for (int c = 0; c < C1V; ++c) hv[c] = h1[(size_t)n * C1V + c];
  #pragma unroll
  for (int hh = 0; hh < H; ++hh) {
    float l = b2l[hh], r = b2r[hh];
    #pragma unroll
    for (int c = 0; c < C1V; ++c) {
      l = fmaf(hv[c], w2l[c * H + hh], l);
      r = fmaf(hv[c], w2r[c * H + hh], r);
    }
    xl2[(size_t)n * H + hh] = l;
    xr2[(size_t)n * H + hh] = r;
  }
}

__global__ __launch_bounds__(256) void k_fin2(
    const float* __restrict__ outnum, const float* __restrict__ denom,
    const float* __restrict__ bias2, float* __restrict__ out) {
  const int n = blockIdx.x * blockDim.x + threadIdx.x;
  if (n >= N_NODES) return;
  float s = 0.f;
  #pragma unroll
  for (int hh = 0; hh < H; ++hh) {
    const float dn = denom[(size_t)n * H + hh];
    s += dn > 0.f ? outnum[(size_t)n * H + hh] / dn : 0.f;
  }
  const float v = s * (1.f / H) + bias2[0];
  out[n] = 1.f / (1.f + __expf(-v));
}

// ---------------------------------------------------------------------------
extern "C" void kernel_launch(void* const* d_in, const int* in_sizes, int n_in,
                              void* d_out, int out_size, void* d_ws,
                              size_t ws_size, hipStream_t stream) {
  const float* x      = (const float*)d_in[0];
  const int*   eidx   = (const int*)d_in[1];
  const float* eattr  = (const float*)d_in[2];
  const int*   batch  = (const int*)d_in[3];
  const float* w1_l   = (const float*)d_in[4];
  const float* b1_l   = (const float*)d_in[5];
  const float* w1_r   = (const float*)d_in[6];
  const float* b1_r   = (const float*)d_in[7];
  const float* w1_e   = (const float*)d_in[8];
  const float* att1   = (const float*)d_in[9];
  const float* bias1  = (const float*)d_in[10];
  const float* gn_w   = (const float*)d_in[11];
  const float* gn_b   = (const float*)d_in[12];
  const float* gn_ms  = (const float*)d_in[13];
  const float* w2_l   = (const float*)d_in[14];
  const float* b2_l   = (const float*)d_in[15];
  const float* w2_r   = (const float*)d_in[16];
  const float* b2_r   = (const float*)d_in[17];
  const float* w2_e   = (const float*)d_in[18];
  const float* att2   = (const float*)d_in[19];
  const float* bias2  = (const float*)d_in[20];
  float* out = (float*)d_out;

  // -------- workspace carve-up (256B aligned) --------
  char* ws = (char*)d_ws;
  size_t off = 0;
  auto carve = [&](size_t bytes) {
    char* p = ws + off;
    off = (off + bytes + 255) & ~(size_t)255;
    return p;
  };
  float*    xl1     = (float*)carve((size_t)N_NODES * H * C1V * 4);  // 20 MB
  float*    xr1     = (float*)carve((size_t)N_NODES * H * C1V * 4);  // 20 MB
  unsigned* lmax1   = (unsigned*)carve((size_t)N_NODES * H * 4);     //  4 MB
  float*    denom1  = (float*)carve((size_t)N_NODES * H * 4);        //  4 MB
  float*    outnum1 = (float*)carve((size_t)N_NODES * H * C1V * 4);  // 20 MB
  float*    y1      = (float*)carve((size_t)N_NODES * C1V * 4);      //  2 MB
  float*    h1      = (float*)carve((size_t)N_NODES * C1V * 4);      //  2 MB
  float*    gn_sum  = (float*)carve(G * C1V * 4);
  float*    gn_cnt  = (float*)carve(G * 4);
  float*    gn_var  = (float*)carve(G * C1V * 4);
  float*    gn_mean = (float*)carve(G * C1V * 4);
  float*    gn_rs   = (float*)carve(G * C1V * 4);
  float*    xl2     = (float*)carve((size_t)N_NODES * H * 4);        //  4 MB
  float*    xr2     = (float*)carve((size_t)N_NODES * H * 4);        //  4 MB
  unsigned* lmax2   = (unsigned*)carve((size_t)N_NODES * H * 4);     //  4 MB
  float*    denom2  = (float*)carve((size_t)N_NODES * H * 4);        //  4 MB
  float*    outnum2 = (float*)carve((size_t)N_NODES * H * 4);        //  4 MB

  // zero-init all accumulators every call (graph-capture safe)
  hipMemsetAsync(lmax1, 0, (size_t)N_NODES * H * 4, stream);
  hipMemsetAsync(denom1, 0, (size_t)N_NODES * H * 4, stream);
  hipMemsetAsync(outnum1, 0, (size_t)N_NODES * H * C1V * 4, stream);
  hipMemsetAsync(gn_sum, 0, G * C1V * 4, stream);
  hipMemsetAsync(gn_cnt, 0, G * 4, stream);
  hipMemsetAsync(gn_var, 0, G * C1V * 4, stream);
  hipMemsetAsync(lmax2, 0, (size_t)N_NODES * H * 4, stream);
  hipMemsetAsync(denom2, 0, (size_t)N_NODES * H * 4, stream);
  hipMemsetAsync(outnum2, 0, (size_t)N_NODES * H * 4, stream);

  const int nodeBlocks = (N_NODES + 255) / 256;
  const int edgeBlocks = (N_EDGES + 255) / 256;
  const int gemmBlocks = (N_NODES + 15) / 16;  // 6250, exact

  // ---- layer 1 ----
  k_node_gemm1<<<gemmBlocks, 128, 0, stream>>>(x, w1_l, b1_l, w1_r, b1_r,
                                               xl1, xr1);
  k_edge_max<C1V><<<edgeBlocks, 256, 0, stream>>>(eidx, eattr, xl1, xr1, w1_e,
                                                  att1, lmax1);
  k_edge_acc<C1V><<<edgeBlocks, 256, 0, stream>>>(eidx, eattr, xl1, xr1, w1_e,
                                                  att1, lmax1, denom1, outnum1);
  k_fin1<<<nodeBlocks, 256, 0, stream>>>(outnum1, denom1, bias1, batch, y1,
                                         gn_sum, gn_cnt);
  // ---- GraphNorm + relu ----
  k_gn_mean<<<(G * C1V + 63) / 64, 64, 0, stream>>>(gn_sum, gn_cnt, gn_mean);
  k_gn_center<<<nodeBlocks, 256, 0, stream>>>(batch, gn_mean, gn_ms, y1,
                                              gn_var);
  k_gn_rstd<<<(G * C1V + 63) / 64, 64, 0, stream>>>(gn_var, gn_cnt, gn_rs);
  k_gn_apply<<<nodeBlocks, 256, 0, stream>>>(batch, y1, gn_rs, gn_w, gn_b, h1);
  // ---- layer 2 ----
  k_lin2<<<nodeBlocks, 256, 0, stream>>>(h1, w2_l, b2_l, w2_r, b2_r, xl2, xr2);
  k_edge_max<1><<<edgeBlocks, 256, 0, stream>>>(eidx, eattr, xl2, xr2, w2_e,
                                                att2, lmax2);
  k_edge_acc<1><<<edgeBlocks, 256, 0, stream>>>(eidx, eattr, xl2, xr2, w2_e,
                                                att2, lmax2, denom2, outnum2);
  k_fin2<<<nodeBlocks, 256, 0, stream>>>(outnum2, denom2, bias2, out);
}